// GeometricGCN_58420145160633
// MI455X (gfx1250) — compile-verified
//
#include <hip/hip_runtime.h>
#include <math.h>

#define NNODES 100000
#define NEDGES 1600000
#define NFEAT  128
#define NEMBED 64
#define NCLS   40

typedef __attribute__((ext_vector_type(2))) float v2f;
typedef __attribute__((ext_vector_type(8))) float v8f;

// ---------------- degree / normalization ----------------

__global__ void deg_init_kernel(float* __restrict__ deg, int n) {
    int i = blockIdx.x * blockDim.x + threadIdx.x;
    if (i < n) deg[i] = 1.0f;  // self-loop contributes 1
}

__global__ void deg_accum_kernel(const int* __restrict__ col, float* __restrict__ deg, int e) {
    int i = blockIdx.x * blockDim.x + threadIdx.x;
    if (i < e) atomicAdd(&deg[col[i]], 1.0f);
}

__global__ void dinv_kernel(float* __restrict__ deg, int n) {
    int i = blockIdx.x * blockDim.x + threadIdx.x;
    if (i < n) deg[i] = rsqrtf(deg[i]);  // deg >= 1 always (self-loops)
}

// ---------------- WMMA fp32 GEMM: C[M,NCOLS] = A[M,K] * B[K,NCOLS] ----------------
// One wave computes a 16-row strip across NT column tiles of 16.
// V_WMMA_F32_16X16X4_F32 layouts (ISA 7.12.2):
//   A 16x4 : lanes 0-15 M=lane (K=k+0,k+1 in v0,v1); lanes 16-31 M=lane-16 (K=k+2,k+3)
//   B 4x16 : mirrored — lanes 0-15 N=lane (K rows k+0,k+1); lanes 16-31 (k+2,k+3)
//   C 16x16: VGPR v, lane l -> M = v + 8*(l>=16), N = l&15

template <int K, int NT, int NCOLS>
__global__ void gemm_wmma_kernel(const float* __restrict__ A, const float* __restrict__ B,
                                 float* __restrict__ C, int M) {
    const int lane    = threadIdx.x & 31;
    const int wave    = threadIdx.x >> 5;
    const int rowTile = blockIdx.x * (blockDim.x >> 5) + wave;
    if (rowTile * 16 >= M) return;               // wave-uniform

    const int row0 = rowTile * 16;
    const int m15  = lane & 15;
    const int half = lane >> 4;
    const int koff = half * 2;

    v8f zero = {};
    v8f acc[NT];
#pragma unroll
    for (int t = 0; t < NT; ++t) acc[t] = zero;

    const float* arow = A + (size_t)(row0 + m15) * K;

    for (int k = 0; k < K; k += 4) {
        v2f a;
        a.x = arow[k + koff];
        a.y = arow[k + koff + 1];
#pragma unroll
        for (int t = 0; t < NT; ++t) {
            const int coln = t * 16 + m15;
            v2f b;
            if (coln < NCOLS) {                  // per-lane select, no effect on WMMA EXEC
                b.x = B[(size_t)(k + koff)     * NCOLS + coln];
                b.y = B[(size_t)(k + koff + 1) * NCOLS + coln];
            } else {
                b.x = 0.0f; b.y = 0.0f;
            }
            acc[t] = __builtin_amdgcn_wmma_f32_16x16x4_f32(
                false, a, false, b, (short)0, acc[t], false, false);
        }
    }

#pragma unroll
    for (int t = 0; t < NT; ++t) {
        const int coln = t * 16 + m15;
        if (coln < NCOLS) {
#pragma unroll
            for (int v = 0; v < 8; ++v) {
                const int r = row0 + v + 8 * half;
                C[(size_t)r * NCOLS + coln] = acc[t][v];
            }
        }
    }
}

// ---------------- GCN aggregation ----------------

// out[i,:] = dinv[i]^2 * src[i,:]   (self-loop term; also fully initializes out)
__global__ void self_loop_init_kernel(const float* __restrict__ src,
                                      const float* __restrict__ dinv,
                                      float* __restrict__ out, int n, int d) {
    int i = blockIdx.x * blockDim.x + threadIdx.x;
    if (i >= n * d) return;
    int node = i / d;
    float dv = dinv[node];
    out[i] = src[i] * dv * dv;
}

// one wave per edge: out[col,:] += dinv[row]*dinv[col] * src[row,:]
__global__ void edge_scatter_kernel(const float* __restrict__ src,
                                    const int* __restrict__ row,
                                    const int* __restrict__ col,
                                    const float* __restrict__ dinv,
                                    float* __restrict__ out, int e, int d) {
    int wave = (blockIdx.x * blockDim.x + threadIdx.x) >> 5;
    int lane = threadIdx.x & 31;
    if (wave >= e) return;                       // wave-uniform
    int r = row[wave];
    int c = col[wave];
    float norm = dinv[r] * dinv[c];
    const float* s = src + (size_t)r * d;
    float* o = out + (size_t)c * d;
    for (int f = lane; f < d; f += 32)
        atomicAdd(&o[f], s[f] * norm);
}

__global__ void bias_relu_kernel(float* __restrict__ x, const float* __restrict__ b,
                                 int n, int d) {
    int i = blockIdx.x * blockDim.x + threadIdx.x;
    if (i >= n * d) return;
    x[i] = fmaxf(x[i] + b[i % d], 0.0f);
}

// one wave per node: logits += bias, then log_softmax over NCLS=40 classes
__global__ void bias_logsoftmax_kernel(float* __restrict__ out, const float* __restrict__ b,
                                       int n, int d) {
    int node = (blockIdx.x * blockDim.x + threadIdx.x) >> 5;
    int lane = threadIdx.x & 31;
    if (node >= n) return;                       // wave-uniform
    float* o = out + (size_t)node * d;
    float v0 = o[lane] + b[lane];                             // lanes 0..31 valid (d=40>32)
    float v1 = (lane + 32 < d) ? (o[lane + 32] + b[lane + 32]) : -INFINITY;
    float m = fmaxf(v0, v1);
    for (int off = 16; off > 0; off >>= 1) m = fmaxf(m, __shfl_xor(m, off, 32));
    float s = expf(v0 - m) + expf(v1 - m);       // exp(-inf)=0 for invalid slots
    for (int off = 16; off > 0; off >>= 1) s += __shfl_xor(s, off, 32);
    float ls = logf(s);
    o[lane] = v0 - m - ls;
    if (lane + 32 < d) o[lane + 32] = v1 - m - ls;
}

// ---------------- launch ----------------

static inline int nblk(long long total, int b) { return (int)((total + b - 1) / b); }

extern "C" void kernel_launch(void* const* d_in, const int* in_sizes, int n_in,
                              void* d_out, int out_size, void* d_ws, size_t ws_size,
                              hipStream_t stream) {
    const float* x   = (const float*)d_in[0];          // [N, 128]
    const int*   ei  = (const int*)d_in[1];            // [2, E]
    const float* W1  = (const float*)d_in[2];          // [128, 64]
    const float* b1  = (const float*)d_in[3];          // [64]
    const float* W2  = (const float*)d_in[4];          // [64, 40]
    const float* b2  = (const float*)d_in[5];          // [40]
    float*       out = (float*)d_out;                  // [N, 40]

    const int N = NNODES, E = NEDGES;
    const int* row = ei;
    const int* col = ei + E;

    float* ws   = (float*)d_ws;
    float* dinv = ws;                                  // N
    float* xw   = ws + N;                              // N*64 (layer1 xW1; reused as layer2 hW2)
    float* h    = ws + N + (size_t)N * NEMBED;         // N*64 (agg1 -> relu in place)

    // --- degrees / symmetric norm ---
    deg_init_kernel<<<nblk(N, 256), 256, 0, stream>>>(dinv, N);
    deg_accum_kernel<<<nblk(E, 256), 256, 0, stream>>>(col, dinv, E);
    dinv_kernel<<<nblk(N, 256), 256, 0, stream>>>(dinv, N);

    // --- layer 1: xw = x @ W1 (WMMA fp32) ---
    {
        int rowTiles = (N + 15) / 16;                  // 6250
        gemm_wmma_kernel<NFEAT, NEMBED / 16, NEMBED>
            <<<nblk(rowTiles, 8), 256, 0, stream>>>(x, W1, xw, N);
    }
    // --- aggregate: h = D^-1/2 A D^-1/2 xw ---
    self_loop_init_kernel<<<nblk((long long)N * NEMBED, 256), 256, 0, stream>>>(xw, dinv, h, N, NEMBED);
    edge_scatter_kernel<<<nblk((long long)E * 32, 256), 256, 0, stream>>>(xw, row, col, dinv, h, E, NEMBED);
    bias_relu_kernel<<<nblk((long long)N * NEMBED, 256), 256, 0, stream>>>(h, b1, N, NEMBED);

    // --- layer 2: xw = h @ W2 (WMMA fp32, 3 column tiles, cols 40..47 masked) ---
    {
        int rowTiles = (N + 15) / 16;
        gemm_wmma_kernel<NEMBED, 3, NCLS>
            <<<nblk(rowTiles, 8), 256, 0, stream>>>(h, W2, xw, N);
    }
    // --- aggregate into d_out ---
    self_loop_init_kernel<<<nblk((long long)N * NCLS, 256), 256, 0, stream>>>(xw, dinv, out, N, NCLS);
    edge_scatter_kernel<<<nblk((long long)E * 32, 256), 256, 0, stream>>>(xw, row, col, dinv, out, E, NCLS);

    // --- bias + log_softmax ---
    bias_logsoftmax_kernel<<<nblk((long long)N * 32, 256), 256, 0, stream>>>(out, b2, N, NCLS);
}